// PEQAug_50105088475749
// MI455X (gfx1250) — compile-verified
//
#include <hip/hip_runtime.h>
#include <math.h>

// ---------------- problem constants ----------------
#define T_LEN    441000
#define NROWS    32
#define NFILT    3
#define NS       8                    // DF-I cascade state dim: x1,x2, y1',y1'', y2',y2'', y3',y3''
#define SSTRIDE  (NS + 1)             // LDS stride 9 (gcd(9,64)=1 -> conflict free)
#define CHUNK    64                   // samples per thread
#define NTHREADS 256
#define WGSAMP   (CHUNK * NTHREADS)   // 16384 samples per workgroup chunk
#define NWG_ROW  ((T_LEN + WGSAMP - 1) / WGSAMP)   // 27
#define TILE_STRIDE (CHUNK + 1)       // 65 dwords per thread (TDM pads 1 dword / 64)
#define TILE_DWORDS (NTHREADS * TILE_STRIDE)       // 16640
#define NMAT     13                   // A^(64*2^k), k=0..12

static_assert(NWG_ROW <= 32, "K2 single-wave scan requires <=32 WG chunks per row");

// ---------------- workspace layout (float offsets) ----------------
#define WS_COEF 0        // 16 floats (15 used)
#define WS_MATS 16       // 13 * 64 floats
#define WS_AGG  1024     // NROWS*NWG_ROW*NS = 6912
#define WS_IN   8192     // 6912

typedef unsigned int       u32;
typedef unsigned long long u64;
typedef __attribute__((ext_vector_type(4))) u32 v4u;
typedef __attribute__((ext_vector_type(8))) int v8i;
typedef __attribute__((ext_vector_type(4))) int v4i;
typedef __attribute__((ext_vector_type(2))) float v2f;
typedef __attribute__((ext_vector_type(8))) float v8f;

// ---------------- DF-I cascade single-sample step (matches reference) ----------------
// c[5f+0..4] = b0,b1,b2,a1,a2 (normalized by a0) for filter f
__device__ __forceinline__ float step8(float s[NS], float x, const float* __restrict__ c)
{
    float y1 = c[0]  * x  + c[1]  * s[0] + c[2]  * s[1] - c[3]  * s[2] - c[4]  * s[3];
    float y2 = c[5]  * y1 + c[6]  * s[2] + c[7]  * s[3] - c[8]  * s[4] - c[9]  * s[5];
    float y3 = c[10] * y2 + c[11] * s[4] + c[12] * s[5] - c[13] * s[6] - c[14] * s[7];
    s[1] = s[0]; s[0] = x;
    s[3] = s[2]; s[2] = y1;
    s[5] = s[4]; s[4] = y2;
    s[7] = s[6]; s[6] = y3;
    return y3;
}

// ---------------- TDM: stage one contiguous tile of floats into LDS ----------------
// Wave 0 only (TDM ignores EXEC -> must be branched around, scalar-uniform).
// D# per CDNA5 ISA: 1-row tile, tensor_dim0 clamped to row
// remainder (OOB reads return zero), pad_enable: 1 dword pad every 64 dwords
// -> LDS layout stride = 65 dwords per 64-sample chunk (bank-conflict free).
__device__ __forceinline__ void tdm_load_tile(float* lds_tile, const float* gsrc, int valid)
{
    const unsigned wave = (unsigned)__builtin_amdgcn_readfirstlane((int)(threadIdx.x >> 5));
    if (wave == 0) {
        const u32 lds_addr = (u32)(u64)(uintptr_t)lds_tile;   // low 32 bits = LDS offset
        const u64 ga = (u64)(uintptr_t)gsrc;
        v4u g0;
        g0[0] = 1u;                                           // count=1, user mode
        g0[1] = lds_addr;
        g0[2] = (u32)(ga & 0xFFFFFFFFull);
        g0[3] = (u32)((ga >> 32) & 0x01FFFFFFull) | (2u << 30); // addr[56:32] | type=2
        const u32 td0 = (u32)valid;
        v8i g1;
        g1[0] = (int)((2u << 16)     // data_size = 4 bytes
                    | (1u << 20)     // pad_enable
                    | (5u << 22));   // pad_interval: every 64 dwords; pad_amount field 0 (1 dword)
        g1[1] = (int)((td0 & 0xFFFFu) << 16);                 // tensor_dim0[15:0]
        g1[2] = (int)(((td0 >> 16) & 0xFFFFu) | (1u << 16));  // tensor_dim0[31:16] | tensor_dim1=1
        g1[3] = (int)(((u32)WGSAMP) << 16);                   // tile_dim0 = 16384
        g1[4] = 1;                                            // tile_dim1 = 1
        g1[5] = (int)WGSAMP;                                  // tensor_dim0_stride
        g1[6] = 0;
        g1[7] = 0;
        v4i z4; z4[0] = z4[1] = z4[2] = z4[3] = 0;
        v8i z8; z8[0] = z8[1] = z8[2] = z8[3] = z8[4] = z8[5] = z8[6] = z8[7] = 0;
        __builtin_amdgcn_tensor_load_to_lds(g0, g1, z4, z4, z8, 0);
        __builtin_amdgcn_s_wait_tensorcnt(0);
    }
}

// ---------------- workgroup-wide inclusive affine scan (VALU; hot path) ----------------
// s_t <- s_t + A^(64*d) * s_{t-d}, d = 1,2,...,128.  Matrices constant (in LDS).
// Caller must have written S[t] = s and issued __syncthreads() beforehand.
__device__ __forceinline__ void scan256(float s[NS], float (*S)[SSTRIDE], const float* mats)
{
    const int t = (int)threadIdx.x;
#pragma unroll 1
    for (int k = 0; k < 8; ++k) {
        const int d = 1 << k;
        float r[NS];
        const bool has = (t >= d);
        if (has) {
#pragma unroll
            for (int i = 0; i < NS; ++i) r[i] = S[t - d][i];
        }
        __syncthreads();
        if (has) {
            const float* M = mats + k * NS * NS;
#pragma unroll
            for (int i = 0; i < NS; ++i) {
                float acc = s[i];
#pragma unroll
                for (int j = 0; j < NS; ++j) acc += M[i * NS + j] * r[j];
                s[i] = acc;
            }
#pragma unroll
            for (int i = 0; i < NS; ++i) S[t][i] = s[i];
        }
        __syncthreads();
    }
}

// ---------------- K0: coefficients + state matrix powers ----------------
__device__ void matsq8(double* M, double* T)
{
    for (int i = 0; i < NS; ++i)
        for (int j = 0; j < NS; ++j) {
            double acc = 0.0;
            for (int k = 0; k < NS; ++k) acc += M[i * NS + k] * M[k * NS + j];
            T[i * NS + j] = acc;
        }
    for (int i = 0; i < NS * NS; ++i) M[i] = T[i];
}

__global__ void peq_k0(const float* __restrict__ qs, const float* __restrict__ gains,
                       float* __restrict__ ws)
{
    if (threadIdx.x != 0) return;
    float c[15];
    for (int f = 0; f < NFILT; ++f) {
        float cf    = 100.0f * powf(30.0f, (float)f * (1.0f / 3.0f)); // 100*(3000/100)^(f/3)
        float w0    = 6.28318530717958647692f * cf / 44100.0f;
        float cw    = cosf(w0), sw = sinf(w0);
        float alpha = sw / (2.0f * qs[f]);
        float Aa    = powf(10.0f, gains[f] * 0.025f);                 // 10^(gain/40)
        float b0 = 1.0f + alpha * Aa;
        float b1 = -2.0f * cw;
        float b2 = 1.0f - alpha * Aa;
        float a0 = 1.0f + alpha / Aa;
        float a2 = 1.0f - alpha / Aa;
        float inv = 1.0f / a0;
        c[5 * f + 0] = b0 * inv;
        c[5 * f + 1] = b1 * inv;
        c[5 * f + 2] = b2 * inv;
        c[5 * f + 3] = b1 * inv;   // a1 == b1
        c[5 * f + 4] = a2 * inv;
    }
    for (int i = 0; i < 15; ++i) ws[WS_COEF + i] = c[i];
    ws[WS_COEF + 15] = 0.0f;

    // Build A (8x8) by probing step8 with basis states, x=0.
    double A[NS * NS];
    for (int j = 0; j < NS; ++j) {
        float st[NS];
        for (int i = 0; i < NS; ++i) st[i] = (i == j) ? 1.0f : 0.0f;
        (void)step8(st, 0.0f, c);
        for (int i = 0; i < NS; ++i) A[i * NS + j] = (double)st[i];
    }
    // B_0 = A^64, B_k = B_{k-1}^2  (f64 squarings, stored as f32)
    double M[NS * NS], T[NS * NS];
    for (int i = 0; i < NS * NS; ++i) M[i] = A[i];
    for (int k = 0; k < 6; ++k) matsq8(M, T);
    for (int m = 0; m < NMAT; ++m) {
        for (int i = 0; i < NS * NS; ++i) ws[WS_MATS + m * NS * NS + i] = (float)M[i];
        if (m < NMAT - 1) matsq8(M, T);
    }
}

// ---------------- K1: per-WG-chunk aggregate (zero-init state after 16384 samples) ----------------
__global__ void __launch_bounds__(NTHREADS) peq_k1(const float* __restrict__ audio,
                                                   float* __restrict__ ws)
{
    extern __shared__ float smem[];
    float* tile = smem;                                  // TILE_DWORDS
    float* mats = smem + TILE_DWORDS;                    // 8*64
    float* coef = mats + 8 * NS * NS;                    // 16
    float (*S)[SSTRIDE] = (float (*)[SSTRIDE])(coef + 16);

    const int t   = (int)threadIdx.x;
    const int wg  = (int)blockIdx.x;
    const int row = (int)blockIdx.y;
    const long long rowBase = (long long)row * T_LEN;
    const int t0    = wg * WGSAMP;
    const int remain = T_LEN - t0;
    const int valid  = remain < WGSAMP ? remain : WGSAMP;

    for (int i = t; i < 8 * NS * NS; i += NTHREADS) mats[i] = ws[WS_MATS + i];
    if (t < 16) coef[t] = ws[WS_COEF + t];
    tdm_load_tile(tile, audio + rowBase + t0, valid);
    __syncthreads();

    float c[15];
#pragma unroll
    for (int i = 0; i < 15; ++i) c[i] = coef[i];

    float s[NS];
#pragma unroll
    for (int i = 0; i < NS; ++i) s[i] = 0.0f;
    const float* xt = tile + t * TILE_STRIDE;
#pragma unroll 4
    for (int i = 0; i < CHUNK; ++i) (void)step8(s, xt[i], c);

#pragma unroll
    for (int i = 0; i < NS; ++i) S[t][i] = s[i];
    __syncthreads();
    scan256(s, S, mats);

    if (t == NTHREADS - 1) {
        float* dst = ws + WS_AGG + (size_t)(row * NWG_ROW + wg) * NS;
#pragma unroll
        for (int i = 0; i < NS; ++i) dst[i] = s[i];
    }
}

// ---------------- K2: per-row scan over WG aggregates, combine via V_WMMA_F32_16X16X4_F32 ----
// Per step: P = M_k x R  with M_k = A^(16384*2^k) (8x8, zero-padded to 16 rows),
// R = 8x32 matrix of shifted state columns (columns with col<d masked to zero,
// so the update is unconditional and EXEC stays all-ones around the WMMAs).
// K=8 contraction = two chained 16x16x4 WMMAs; two column groups cover 32 elements.
__global__ void __launch_bounds__(32) peq_k2(float* __restrict__ ws)
{
    __shared__ float S2[32][SSTRIDE];
    __shared__ float P[32][SSTRIDE];
    const int row = (int)blockIdx.x;
    const int t   = (int)threadIdx.x;
    const int lo  = t & 15;     // A: output row / B: column within group / D: column
    const int hi  = t >> 4;     // selects K pair {0,1} (lanes 0-15) vs {2,3} (lanes 16-31)

    float s[NS];
#pragma unroll
    for (int i = 0; i < NS; ++i) s[i] = 0.0f;
    if (t >= 1 && t < NWG_ROW) {                   // shift trick: v'_j = agg_{j-1}, v'_0 = 0
        const float* src = ws + WS_AGG + (size_t)(row * NWG_ROW + t - 1) * NS;
#pragma unroll
        for (int i = 0; i < NS; ++i) s[i] = src[i];
    }
#pragma unroll
    for (int i = 0; i < NS; ++i) S2[t][i] = s[i];
    __syncthreads();

#pragma unroll 1
    for (int k = 0; k < 5; ++k) {
        const int d = 1 << k;
        const float* M = ws + WS_MATS + (8 + k) * NS * NS;   // A^(16384*2^k), 8x8 row-major
        // A-matrix operands (16x4 f32 layout: lane = row; VGPR0/1 = K {0,1} lo-lanes, {2,3} hi-lanes)
        const float am   = (lo < NS) ? 1.0f : 0.0f;          // zero-pad rows 8..15
        const int   rbas = (lo < NS ? lo : 0) * NS;
        v2f a1, a2;
        a1.x = am * M[rbas + 2 * hi + 0];
        a1.y = am * M[rbas + 2 * hi + 1];
        a2.x = am * M[rbas + 4 + 2 * hi + 0];                // K = 4..7 slice
        a2.y = am * M[rbas + 4 + 2 * hi + 1];
#pragma unroll
        for (int g = 0; g < 2; ++g) {
            const int   col = g * 16 + lo;
            const float bm  = (col >= d) ? 1.0f : 0.0f;      // mask columns without a predecessor
            const int   sj  = (col >= d) ? (col - d) : 0;
            v2f b1, b2;                                       // B 4x16: lane = column; K pairs as A
            b1.x = bm * S2[sj][2 * hi + 0];
            b1.y = bm * S2[sj][2 * hi + 1];
            b2.x = bm * S2[sj][4 + 2 * hi + 0];
            b2.y = bm * S2[sj][4 + 2 * hi + 1];
            v8f acc = {};
            acc = __builtin_amdgcn_wmma_f32_16x16x4_f32(false, a2, false, b2, (short)0, acc, false, false);
            acc = __builtin_amdgcn_wmma_f32_16x16x4_f32(false, a1, false, b1, (short)0, acc, false, false);
            // D rows 0..7 of column (g*16+n) live in lane n (0..15), VGPRs 0..7
            if (t < 16) {
#pragma unroll
                for (int i = 0; i < NS; ++i) P[g * 16 + t][i] = acc[i];
            }
        }
        __syncthreads();
#pragma unroll
        for (int i = 0; i < NS; ++i) { s[i] += P[t][i]; S2[t][i] = s[i]; }
        __syncthreads();
    }
    if (t < NWG_ROW) {
        float* dst = ws + WS_IN + (size_t)(row * NWG_ROW + t) * NS;
#pragma unroll
        for (int i = 0; i < NS; ++i) dst[i] = s[i];
    }
}

// ---------------- K3: recover exact per-thread incoming states, replay, write output ----------------
__global__ void __launch_bounds__(NTHREADS) peq_k3(const float* __restrict__ audio,
                                                   const float* __restrict__ ws,
                                                   float* __restrict__ out)
{
    extern __shared__ float smem[];
    float* tile = smem;
    float* mats = smem + TILE_DWORDS;
    float* coef = mats + 8 * NS * NS;
    float (*S)[SSTRIDE] = (float (*)[SSTRIDE])(coef + 16);

    const int t   = (int)threadIdx.x;
    const int wg  = (int)blockIdx.x;
    const int row = (int)blockIdx.y;
    const long long rowBase = (long long)row * T_LEN;
    const int t0    = wg * WGSAMP;
    const int remain = T_LEN - t0;
    const int valid  = remain < WGSAMP ? remain : WGSAMP;

    for (int i = t; i < 8 * NS * NS; i += NTHREADS) mats[i] = ws[WS_MATS + i];
    if (t < 16) coef[t] = ws[WS_COEF + t];
    tdm_load_tile(tile, audio + rowBase + t0, valid);
    __syncthreads();

    float c[15];
#pragma unroll
    for (int i = 0; i < 15; ++i) c[i] = coef[i];

    // pass A: per-thread chunk transform vector v (zero-init run)
    float v[NS];
#pragma unroll
    for (int i = 0; i < NS; ++i) v[i] = 0.0f;
    float* xt = tile + t * TILE_STRIDE;
#pragma unroll 4
    for (int i = 0; i < CHUNK; ++i) (void)step8(v, xt[i], c);

    // shift trick: element_t = v_{t-1}, element_0 = incoming WG state
#pragma unroll
    for (int i = 0; i < NS; ++i) S[t][i] = v[i];
    __syncthreads();
    float vp[NS];
    if (t == 0) {
        const float* src = ws + WS_IN + (size_t)(row * NWG_ROW + wg) * NS;
#pragma unroll
        for (int i = 0; i < NS; ++i) vp[i] = src[i];
    } else {
#pragma unroll
        for (int i = 0; i < NS; ++i) vp[i] = S[t - 1][i];
    }
    __syncthreads();
#pragma unroll
    for (int i = 0; i < NS; ++i) S[t][i] = vp[i];
    __syncthreads();
    scan256(vp, S, mats);          // vp = exact incoming state for this thread's chunk

    // pass B: replay with correct state, write y back into the LDS tile
#pragma unroll 4
    for (int i = 0; i < CHUNK; ++i) {
        float y = step8(vp, xt[i], c);
        xt[i] = y;
    }
    __syncthreads();

    // coalesced store (de-pad the 65-dword stride)
    for (int idx = t; idx < valid; idx += NTHREADS) {
        const int q = idx >> 6, r = idx & 63;
        out[rowBase + t0 + idx] = tile[q * TILE_STRIDE + r];
    }
}

// ---------------- host launch ----------------
extern "C" void kernel_launch(void* const* d_in, const int* in_sizes, int n_in,
                              void* d_out, int out_size, void* d_ws, size_t ws_size,
                              hipStream_t stream)
{
    (void)in_sizes; (void)n_in; (void)out_size; (void)ws_size;
    const float* audio = (const float*)d_in[0];
    const float* qs    = (const float*)d_in[1];
    const float* gains = (const float*)d_in[2];
    float* out = (float*)d_out;
    float* ws  = (float*)d_ws;

    peq_k0<<<1, 1, 0, stream>>>(qs, gains, ws);

    const size_t smem = (size_t)(TILE_DWORDS + 8 * NS * NS + 16 + NTHREADS * SSTRIDE) * sizeof(float);
    dim3 grid(NWG_ROW, NROWS);
    peq_k1<<<grid, NTHREADS, smem, stream>>>(audio, ws);
    peq_k2<<<NROWS, 32, 0, stream>>>(ws);
    peq_k3<<<grid, NTHREADS, smem, stream>>>(audio, ws, out);
}